// TransformerEncoder_7421703488045
// MI455X (gfx1250) — compile-verified
//
#include <hip/hip_runtime.h>

#define D 512
#define NB 128
#define EPSF 1e-5f

typedef __attribute__((ext_vector_type(2))) float v2f;
typedef __attribute__((ext_vector_type(8))) float v8f;
typedef __attribute__((ext_vector_type(4))) int   v4i;

#if __has_builtin(__builtin_amdgcn_exp2f)
#define EXP2F(x) __builtin_amdgcn_exp2f(x)
#else
#define EXP2F(x) exp2f(x)
#endif

#if __has_builtin(__builtin_amdgcn_global_load_async_to_lds_b128) && \
    __has_builtin(__builtin_amdgcn_s_wait_asynccnt)
#define HAS_ASYNC_LDS 1
#else
#define HAS_ASYNC_LDS 0
#endif

#define AS1 __attribute__((address_space(1)))
#define AS3 __attribute__((address_space(3)))

// ---------------------------------------------------------------------------
// WMMA fp32 GEMM:  C[m,n] = sum_k A[m,k] * W[n,k] + bias[n] (+ res[m,n])
// A: 128x512 row-major, W: 512x512 row-major (W^T is the WMMA B matrix).
// Workgroup = 256 threads = 8 waves arranged 2(M) x 4(N); WG tile 32x64.
// K marched in LDS stages of 32, each stage = 8 x V_WMMA_F32_16X16X4_F32.
// Stages are double-buffered with GLOBAL_LOAD_ASYNC_TO_LDS_B128 (ASYNCcnt)
// so the DMA of stage s+1 overlaps the WMMAs of stage s — critical here since
// only ~1 WG/WGP is resident, so there are no other waves to hide latency.
// ---------------------------------------------------------------------------
#define TM 32
#define TN 64
#define TK 32
#define LDK 36   // padded LDS row stride: 36*4B = 144B (16B-aligned rows for
                 // async b128 chunks) and 36m mod 64 distinct for m=0..15.

__device__ __forceinline__ void gemm_tile(const float* __restrict__ A,
                                          const float* __restrict__ W,
                                          const float* __restrict__ bias,
                                          const float* __restrict__ res,
                                          float* __restrict__ C)
{
  const int K = D, N = D;
  const int tid  = threadIdx.x;
  const int wave = tid >> 5, lane = tid & 31;
  const int wm = wave >> 2, wn = wave & 3;      // 2 x 4 wave grid
  const int lm = lane & 15, lh = lane >> 4;     // lane -> (m/n, half) per ISA layout
  const int m0 = blockIdx.y * TM;
  const int n0 = blockIdx.x * TN;

  v8f acc = {};

#if HAS_ASYNC_LDS
  __shared__ __align__(16) float As[2][TM * LDK];
  __shared__ __align__(16) float Bs[2][TN * LDK];

  // 16B-chunk assignments: A is 32x32 floats = 256 chunks (1/thread),
  // B is 64x32 floats = 512 chunks (2/thread).
  const int ra  = tid >> 3;             // A row 0..31
  const int rb0 = tid >> 3;             // B rows 0..31
  const int rb1 = 32 + (tid >> 3);      // B rows 32..63
  const int cc  = (tid & 7) * 4;        // k-column (floats), 16B aligned

  const float* gA  = A + (size_t)(m0 + ra)  * K + cc;
  const float* gB0 = W + (size_t)(n0 + rb0) * K + cc;
  const float* gB1 = W + (size_t)(n0 + rb1) * K + cc;

#define ISSUE_STAGE(buf, kk)                                                   \
  do {                                                                         \
    __builtin_amdgcn_global_load_async_to_lds_b128(                            \
        (AS1 v4i*)(gA + (kk)),                                                 \
        (AS3 v4i*)&As[(buf)][ra * LDK + cc], 0, 0);                            \
    __builtin_amdgcn_global_load_async_to_lds_b128(                            \
        (AS1 v4i*)(gB0 + (kk)),                                                \
        (AS3 v4i*)&Bs[(buf)][rb0 * LDK + cc], 0, 0);                           \
    __builtin_amdgcn_global_load_async_to_lds_b128(                            \
        (AS1 v4i*)(gB1 + (kk)),                                                \
        (AS3 v4i*)&Bs[(buf)][rb1 * LDK + cc], 0, 0);                           \
  } while (0)

  ISSUE_STAGE(0, 0);
  const int NSTAGE = K / TK;            // 16
  for (int s = 0; s < NSTAGE; ++s) {
    const int buf = s & 1;
    if (s + 1 < NSTAGE) {
      ISSUE_STAGE(buf ^ 1, (s + 1) * TK);        // prefetch next stage (DMA)
      __builtin_amdgcn_s_wait_asynccnt(3);       // current stage's 3 ops done
    } else {
      __builtin_amdgcn_s_wait_asynccnt(0);
    }
    __syncthreads();                             // publish stage to all waves

    const float* ap = &As[buf][(wm * 16 + lm) * LDK + 2 * lh];
    const float* bp = &Bs[buf][(wn * 16 + lm) * LDK + 2 * lh];
    #pragma unroll
    for (int ks = 0; ks < TK / 4; ++ks) {
      v2f a = *(const v2f*)(ap + ks * 4);
      v2f b = *(const v2f*)(bp + ks * 4);
      acc = __builtin_amdgcn_wmma_f32_16x16x4_f32(false, a, false, b,
                                                  (short)0, acc, false, false);
    }
    __syncthreads();   // everyone done reading buf before it is re-issued
  }
#undef ISSUE_STAGE

#else  // ---------- fallback: synchronous staging + hardware prefetch ----------
  __shared__ __align__(16) float As[TM * LDK];
  __shared__ __align__(16) float Bs[TN * LDK];

  for (int kk = 0; kk < K; kk += TK) {
    __syncthreads();
    #pragma unroll
    for (int i = 0; i < (TM * TK) / 256; ++i) {
      int idx = tid + i * 256;
      int r = idx >> 5, c = idx & 31;
      As[r * LDK + c] = A[(m0 + r) * K + kk + c];
      if (kk + TK < K) __builtin_prefetch(&A[(m0 + r) * K + kk + TK + c], 0, 1);
    }
    #pragma unroll
    for (int i = 0; i < (TN * TK) / 256; ++i) {
      int idx = tid + i * 256;
      int n = idx >> 5, c = idx & 31;
      Bs[n * LDK + c] = W[(n0 + n) * K + kk + c];
      if (kk + TK < K) __builtin_prefetch(&W[(n0 + n) * K + kk + TK + c], 0, 1);
    }
    __syncthreads();

    const float* ap = &As[(wm * 16 + lm) * LDK + 2 * lh];
    const float* bp = &Bs[(wn * 16 + lm) * LDK + 2 * lh];
    #pragma unroll
    for (int ks = 0; ks < TK / 4; ++ks) {
      v2f a = *(const v2f*)(ap + ks * 4);
      v2f b = *(const v2f*)(bp + ks * 4);
      acc = __builtin_amdgcn_wmma_f32_16x16x4_f32(false, a, false, b,
                                                  (short)0, acc, false, false);
    }
  }
#endif

  // Epilogue: C vgpr r -> m = r + 8*lh, n = lm  (ISA 16x16 f32 C layout)
  const int n = n0 + wn * 16 + lm;
  const float bv = bias[n];
  #pragma unroll
  for (int r = 0; r < 8; ++r) {
    int m = m0 + wm * 16 + r + 8 * lh;
    float v = acc[r] + bv;
    if (res) v += res[m * N + n];
    C[m * N + n] = v;
  }
}

// z = which*3 + ch : which 0 -> w_k = h@Wk^T+bk ; which 1 -> w_q = kidx@Wq^T+bq
__global__ __launch_bounds__(256) void proj_kernel(
    const float* h_r, const float* h_g, const float* h_b,
    const float* k_r, const float* k_g, const float* k_b,
    const float* Wk, const float* bk, const float* Wq, const float* bq,
    float* wk_ws, float* wq_ws)
{
  int z = blockIdx.z;
  int which = z / 3, ch = z % 3;
  const float* ins[6] = {h_r, h_g, h_b, k_r, k_g, k_b};
  const float* In   = ins[which * 3 + ch];
  const float* W    = which ? Wq : Wk;
  const float* bias = which ? bq : bk;
  float* Out = (which ? wq_ws : wk_ws) + (size_t)ch * NB * D;
  gemm_tile(In, W, bias, nullptr, Out);
}

__global__ __launch_bounds__(256) void fc_kernel(
    const float* yln_ws, const float* fc_w, const float* fc_b, float* out)
{
  int ch = blockIdx.z;
  const float* In = yln_ws + (size_t)ch * NB * D;
  gemm_tile(In, fc_w, fc_b, In, out + (size_t)ch * NB * D);
}

// ---------------------------------------------------------------------------
// Fused attention core: exploits rank-1 structure of the LN'd outer products.
// One 256-thread WG per (batch, channel); each thread owns 2 of 512 elements.
// ---------------------------------------------------------------------------
template <int OP>  // 0=sum 1=max 2=min
__device__ __forceinline__ float blockReduce(float v, float* s)
{
  int t = threadIdx.x;
  __syncthreads();
  s[t] = v;
  __syncthreads();
  #pragma unroll
  for (int off = 128; off > 0; off >>= 1) {
    if (t < off) {
      float a = s[t], b = s[t + off];
      s[t] = (OP == 0) ? (a + b) : ((OP == 1) ? fmaxf(a, b) : fminf(a, b));
    }
    __syncthreads();
  }
  return s[0];
}

__global__ __launch_bounds__(256) void attn_kernel(
    const float* x_r, const float* x_g, const float* x_b,
    const float* h_r, const float* h_g, const float* h_b,
    const float* k_r, const float* k_g, const float* k_b,
    const float* kn_g, const float* kn_b,
    const float* norm_g, const float* norm_b,
    const float* wk_ws, const float* wq_ws, float* yln_ws)
{
  const int b  = blockIdx.x;
  const int ch = blockIdx.y;
  const float* xs[3] = {x_r, x_g, x_b};
  const float* hs[3] = {h_r, h_g, h_b};
  const float* ks[3] = {k_r, k_g, k_b};
  const float* vin = xs[ch] + (size_t)b * D;   // v row
  const float* kin = hs[ch] + (size_t)b * D;   // k row (from h)
  const float* qin = ks[ch] + (size_t)b * D;   // q row (from kidx)
  const float* wk  = wk_ws + ((size_t)ch * NB + b) * D;
  const float* wq  = wq_ws + ((size_t)ch * NB + b) * D;

  __shared__ float sA[D];    // a_i
  __shared__ float sU[D];    // u_k * log2(e)
  __shared__ float sM[D];    // per-column max (log2 domain)
  __shared__ float sWV[D];   // v_k / Z_k
  __shared__ float sV[D];    // v row
  __shared__ float sRed[256];

  const int t = threadIdx.x;

  float wkv[2], wqv[2], kv[2], qv[2], gv[2], bbv[2];
  float swk = 0.f, swk2 = 0.f, swq = 0.f, swq2 = 0.f;
  #pragma unroll
  for (int j = 0; j < 2; ++j) {
    int i = t + j * 256;
    wkv[j] = wk[i];  wqv[j] = wq[i];
    kv[j]  = kin[i]; qv[j]  = qin[i];
    gv[j]  = kn_g[i]; bbv[j] = kn_b[i];
    sV[i] = vin[i];
    swk += wkv[j]; swk2 += wkv[j] * wkv[j];
    swq += wqv[j]; swq2 += wqv[j] * wqv[j];
  }
  swk  = blockReduce<0>(swk,  sRed);
  swk2 = blockReduce<0>(swk2, sRed);
  swq  = blockReduce<0>(swq,  sRed);
  swq2 = blockReduce<0>(swq2, sRed);
  const float muk  = swk * (1.f / D);
  const float vark = swk2 * (1.f / D) - muk * muk;
  const float muq  = swq * (1.f / D);
  const float varq = swq2 * (1.f / D) - muq * muq;

  // a_i = k_i / sqrt(k_i^2*var_wk + eps) ; a'_j similarly from q.
  // c_j = (w_kj - mu_wk)*gamma_j ; c'_k = (w_qk - mu_wq)*gamma_k.
  float S1 = 0.f, S2 = 0.f, amax = -3.0e38f, amin = 3.0e38f;
  float cpv[2];
  #pragma unroll
  for (int j = 0; j < 2; ++j) {
    int i = t + j * 256;
    float a  = kv[j] * rsqrtf(kv[j] * kv[j] * vark + EPSF);
    float ap = qv[j] * rsqrtf(qv[j] * qv[j] * varq + EPSF);
    sA[i] = a;
    float c = (wkv[j] - muk) * gv[j];
    cpv[j]  = (wqv[j] - muq) * gv[j];
    S1 += c * ap;   // sum_j c_j a'_j
    S2 += c;        // sum_j c_j
    amax = fmaxf(amax, a);
    amin = fminf(amin, a);
  }
  S1   = blockReduce<0>(S1, sRed);
  S2   = blockReduce<0>(S2, sRed);
  amax = blockReduce<1>(amax, sRed);
  amin = blockReduce<2>(amin, sRed);

  // scores_ik = a_i*u_k + t_k; t_k cancels in softmax over i (axis=1).
  const float LOG2E = 1.4426950408889634f;
  #pragma unroll
  for (int j = 0; j < 2; ++j) {
    int i = t + j * 256;
    float u  = cpv[j] * S1 + bbv[j] * S2;
    float ul = u * LOG2E;
    sU[i] = ul;
    sM[i] = fmaxf(ul * amax, ul * amin);   // max_i of a_i*ul (log2 domain)
  }
  __syncthreads();

  // Phase 1: column partition functions Z_k = sum_i 2^(a_i*ul_k - M_k); >= 1.
  #pragma unroll
  for (int j = 0; j < 2; ++j) {
    int kc = t + j * 256;
    float ul = sU[kc], Ml = sM[kc], z = 0.f;
    #pragma unroll 8
    for (int i = 0; i < D; ++i)
      z += EXP2F(fmaf(sA[i], ul, -Ml));
    sWV[kc] = sV[kc] / z;
  }
  __syncthreads();

  // Phase 2: out_i = sum_k 2^(a_i*ul_k - M_k) * (v_k/Z_k);  y = out + v; then LN.
  float y[2], ys = 0.f, ys2 = 0.f;
  #pragma unroll
  for (int j = 0; j < 2; ++j) {
    int ir = t + j * 256;
    float a = sA[ir], acc = 0.f;
    #pragma unroll 8
    for (int k = 0; k < D; ++k)
      acc += EXP2F(fmaf(a, sU[k], -sM[k])) * sWV[k];
    y[j] = acc + sV[ir];
    ys += y[j]; ys2 += y[j] * y[j];
  }
  ys  = blockReduce<0>(ys, sRed);
  ys2 = blockReduce<0>(ys2, sRed);
  const float muy = ys * (1.f / D);
  const float rs  = rsqrtf(ys2 * (1.f / D) - muy * muy + EPSF);

  float* outY = yln_ws + ((size_t)ch * NB + b) * D;
  #pragma unroll
  for (int j = 0; j < 2; ++j) {
    int i = t + j * 256;
    outY[i] = (y[j] - muy) * rs * norm_g[i] + norm_b[i];
  }
}

// ---------------------------------------------------------------------------
extern "C" void kernel_launch(void* const* d_in, const int* in_sizes, int n_in,
                              void* d_out, int out_size, void* d_ws, size_t ws_size,
                              hipStream_t stream)
{
  const float* r    = (const float*)d_in[0];
  const float* g    = (const float*)d_in[1];
  const float* bch  = (const float*)d_in[2];
  const float* h_r  = (const float*)d_in[3];
  const float* h_g  = (const float*)d_in[4];
  const float* h_b  = (const float*)d_in[5];
  const float* k_r  = (const float*)d_in[6];
  const float* k_g  = (const float*)d_in[7];
  const float* k_b  = (const float*)d_in[8];
  const float* Wk   = (const float*)d_in[9];
  const float* bk   = (const float*)d_in[10];
  const float* Wq   = (const float*)d_in[11];
  const float* bq   = (const float*)d_in[12];
  const float* kn_g = (const float*)d_in[13];
  const float* kn_b = (const float*)d_in[14];
  const float* norm_g = (const float*)d_in[15];
  const float* norm_b = (const float*)d_in[16];
  const float* fc_w = (const float*)d_in[17];
  const float* fc_b = (const float*)d_in[18];

  const size_t chunk = (size_t)3 * NB * D;   // floats per ws region
  float* wk_ws  = (float*)d_ws;
  float* wq_ws  = wk_ws + chunk;
  float* yln_ws = wq_ws + chunk;

  dim3 blk(256);
  dim3 gProj(D / TN, NB / TM, 6);   // 8 x 4 x 6
  proj_kernel<<<gProj, blk, 0, stream>>>(h_r, h_g, h_b, k_r, k_g, k_b,
                                         Wk, bk, Wq, bq, wk_ws, wq_ws);

  dim3 gAttn(NB, 3);                // 128 x 3
  attn_kernel<<<gAttn, blk, 0, stream>>>(r, g, bch, h_r, h_g, h_b,
                                         k_r, k_g, k_b, kn_g, kn_b,
                                         norm_g, norm_b, wk_ws, wq_ws, yln_ws);

  dim3 gFc(D / TN, NB / TM, 3);     // 8 x 4 x 3
  fc_kernel<<<gFc, blk, 0, stream>>>(yln_ws, fc_w, fc_b, (float*)d_out);
}